// GNNBackbone_7310034338075
// MI455X (gfx1250) — compile-verified
//
#include <hip/hip_runtime.h>
#include <hip/hip_bf16.h>

// ---------------------------------------------------------------------------
// GAT backbone for MI455X (gfx1250, wave32).
//   h = nf @ W0^T + b0                      GEMM via V_WMMA_F32_16X16X4_F32
//   3x GAT layer:
//     a_src[n] = h[n] . Wattn[:256], a_dst[n] = h[n] . Wattn[256:]   (wave dot)
//     ef[e] = a_src[src] + a_dst[dst]; segment softmax over dst (atomics in L2)
//     agg[dst] += alpha * h[src]            (one wave per edge, f32 atomics)
//     h' = relu([h | agg] @ Wlin^T)         GEMM K=512 via f32 WMMA
// GEMM: one wave computes a 16x64 strip (4 independent accumulator chains,
// 4x A-fragment reuse). Concat-K runs as two branch-free double-buffered
// phases; sched_group_barrier(VMEM-read x5, WMMA x4) enforces prefetch-ahead
// issue order so waitcnts cover a full iteration of latency.
// Working set (~35 MB) is L2-resident (192 MB).
// ---------------------------------------------------------------------------

#define NF_IN  128
#define HIDDEN 256
#define DEPTH  3

typedef __attribute__((ext_vector_type(2))) float v2f;
typedef __attribute__((ext_vector_type(8))) float v8f;

// ---- monotonic float <-> uint for atomicMax-based segment max --------------
__device__ __forceinline__ unsigned fenc(float x) {
    unsigned u = __float_as_uint(x);
    return (u & 0x80000000u) ? ~u : (u | 0x80000000u);
}
__device__ __forceinline__ float fdec(unsigned k) {
    return (k & 0x80000000u) ? __uint_as_float(k & 0x7fffffffu)
                             : __uint_as_float(~k);
}
#define ENC_NEGINF 0x007FFFFFu  // fenc(-inf)

// ---- fills -----------------------------------------------------------------
__global__ void fill_u32_kernel(unsigned* __restrict__ p, unsigned v, int n) {
    int i = blockIdx.x * blockDim.x + threadIdx.x;
    if (i < n) p[i] = v;
}
__global__ void fill_f32_kernel(float* __restrict__ p, float v, int n) {
    int i = blockIdx.x * blockDim.x + threadIdx.x;
    if (i < n) p[i] = v;
}

// ---- fp32 WMMA GEMM:  C[M,N] = act( [A | A2][M,Ktot] @ W[N,Ktot]^T + bias )
// A/A2 row-major with row stride = lda (holding K1 / Ktot-K1 columns resp.).
// W row-major [N, Ktot]. One wave computes one 16x64 output strip.
// Fragment layouts per CDNA5 ISA 7.12.2 (32-bit A 16x4 / B,C,D row-per-VGPR).
#define NSTRIP 4   // 16x16 tiles per wave along N
__global__ __launch_bounds__(256)
void gemm_atb_wmma_kernel(const float* __restrict__ A,
                          const float* __restrict__ A2,
                          int K1, int Ktot, int lda,
                          const float* __restrict__ W,
                          const float* __restrict__ bias,
                          float* __restrict__ C,
                          int M, int N, int relu)
{
    const int lane = threadIdx.x & 31;
    const int wave = threadIdx.x >> 5;
    const int half = lane >> 4;         // 0: lanes 0-15, 1: lanes 16-31
    const int l16  = lane & 15;

    const int stripsN = N >> 6;                     // 64-wide strips
    const int nTiles  = (M >> 4) * stripsN;
    const int tile    = blockIdx.x * (blockDim.x >> 5) + wave;
    if (tile >= nTiles) return;

    const int m0 = (tile / stripsN) << 4;
    const int n0 = (tile % stripsN) << 6;

    v8f acc[NSTRIP] = {};
    if (bias) {
#pragma unroll
        for (int j = 0; j < NSTRIP; ++j) {
            const float b = bias[n0 + j * 16 + l16];
#pragma unroll
            for (int r = 0; r < 8; ++r) acc[j][r] = b;
        }
    }

    const int arow = m0 + l16;
    const int koff = half * 2;          // lane-dependent K sub-offset

    // per-strip weight row base: strip j covers cols n0+16j..n0+16j+15
    const float* __restrict__ Wrow[NSTRIP];
#pragma unroll
    for (int j = 0; j < NSTRIP; ++j)
        Wrow[j] = W + (size_t)(n0 + j * 16 + l16) * Ktot + koff;

    // One concat phase: branch-free double-buffered loop over a single fixed
    // base pointer; last step peeled. kbase = position of this phase in W's K.
    auto runPhase = [&](const float* __restrict__ Abase, int kbase, int Klen) {
        const float* __restrict__ Ap = Abase + (size_t)arow * lda + koff;
        v2f a_cur = *(const v2f*)Ap;
        v2f b_cur[NSTRIP];
#pragma unroll
        for (int j = 0; j < NSTRIP; ++j)
            b_cur[j] = *(const v2f*)(Wrow[j] + kbase);

#pragma unroll 4
        for (int k0 = 0; k0 + 4 < Klen; k0 += 4) {
            const int kn = k0 + 4;
            v2f a_nxt = *(const v2f*)(Ap + kn);   // prefetch next step
            v2f b_nxt[NSTRIP];
#pragma unroll
            for (int j = 0; j < NSTRIP; ++j)
                b_nxt[j] = *(const v2f*)(Wrow[j] + kbase + kn);
#pragma unroll
            for (int j = 0; j < NSTRIP; ++j) {
                acc[j] = __builtin_amdgcn_wmma_f32_16x16x4_f32(
                    false, a_cur, false, b_cur[j], (short)0, acc[j],
                    false, false);
            }
            a_cur = a_nxt;
#pragma unroll
            for (int j = 0; j < NSTRIP; ++j) b_cur[j] = b_nxt[j];
            // enforce issue order: 5 VMEM reads (prefetch k+1), then 4 WMMAs
            // (consuming step k) -> partial s_wait_loadcnt instead of 0.
            __builtin_amdgcn_sched_group_barrier(0x020, 5, 0); // VMEM read
            __builtin_amdgcn_sched_group_barrier(0x008, 4, 0); // MFMA/WMMA
        }
        // peeled final step (no prefetch)
#pragma unroll
        for (int j = 0; j < NSTRIP; ++j) {
            acc[j] = __builtin_amdgcn_wmma_f32_16x16x4_f32(
                false, a_cur, false, b_cur[j], (short)0, acc[j], false, false);
        }
    };

    const int kEnd = (K1 < Ktot) ? K1 : Ktot;
    runPhase(A, 0, kEnd);
    if (A2 && Ktot > K1) runPhase(A2, K1, Ktot - K1);

    // store: acc[j] VGPR r -> row m0 + r + 8*half, col n0 + 16j + l16
    float* __restrict__ Crow = C + (size_t)(m0 + half * 8) * N + n0 + l16;
#pragma unroll
    for (int j = 0; j < NSTRIP; ++j) {
#pragma unroll
        for (int r = 0; r < 8; ++r) {
            float v = acc[j][r];
            if (relu) v = v > 0.0f ? v : 0.0f;
            Crow[(size_t)r * N + j * 16] = v;
        }
    }
}

// ---- per-node attention terms: a_src[n]=h[n].Wa[:256], a_dst=h[n].Wa[256:]
__global__ __launch_bounds__(256)
void att_node_kernel(const float* __restrict__ h,
                     const float* __restrict__ Wa,
                     float* __restrict__ asrc, float* __restrict__ adst,
                     int nNodes)
{
    const int lane = threadIdx.x & 31;
    const int node = (blockIdx.x * blockDim.x + threadIdx.x) >> 5;
    if (node >= nNodes) return;
    const float* __restrict__ hp = h + (size_t)node * HIDDEN;
    float s0 = 0.0f, s1 = 0.0f;
#pragma unroll
    for (int j = 0; j < HIDDEN / 32; ++j) {
        const int c = lane + 32 * j;
        const float v = hp[c];
        s0 += v * Wa[c];
        s1 += v * Wa[HIDDEN + c];
    }
#pragma unroll
    for (int off = 16; off > 0; off >>= 1) {
        s0 += __shfl_down(s0, off, 32);
        s1 += __shfl_down(s1, off, 32);
    }
    if (lane == 0) { asrc[node] = s0; adst[node] = s1; }
}

// ---- per-edge logit + segment max (ordered-uint atomicMax) -----------------
__global__ __launch_bounds__(256)
void edge_logit_max_kernel(const int* __restrict__ src,
                           const int* __restrict__ dst,
                           const float* __restrict__ asrc,
                           const float* __restrict__ adst,
                           float* __restrict__ ef,
                           unsigned* __restrict__ segmax, int E)
{
    int e = blockIdx.x * blockDim.x + threadIdx.x;
    if (e >= E) return;
    const int d = dst[e];
    const float x = asrc[src[e]] + adst[d];
    ef[e] = x;
    atomicMax(&segmax[d], fenc(x));
}

// ---- per-edge exp + segment sum -------------------------------------------
__global__ __launch_bounds__(256)
void edge_exp_sum_kernel(const int* __restrict__ dst,
                         const float* __restrict__ ef,
                         const unsigned* __restrict__ segmax,
                         float* __restrict__ efexp,
                         float* __restrict__ denom, int E)
{
    int e = blockIdx.x * blockDim.x + threadIdx.x;
    if (e >= E) return;
    const int d = dst[e];
    const float w = expf(ef[e] - fdec(segmax[d]));
    efexp[e] = w;
    atomicAdd(&denom[d], w);
}

// ---- weighted scatter-aggregate: one wave32 per edge, 8 floats per lane ----
__global__ __launch_bounds__(256)
void edge_aggregate_kernel(const int* __restrict__ src,
                           const int* __restrict__ dst,
                           const float* __restrict__ h,
                           const float* __restrict__ efexp,
                           const float* __restrict__ denom,
                           float* __restrict__ agg, int E)
{
    const int lane = threadIdx.x & 31;
    const int e    = (blockIdx.x * blockDim.x + threadIdx.x) >> 5;
    if (e >= E) return;
    const int s = src[e];
    const int d = dst[e];
    const float alpha = efexp[e] / denom[d];
    const float* __restrict__ hs = h + (size_t)s * HIDDEN;
    float* __restrict__ ag = agg + (size_t)d * HIDDEN;
#pragma unroll
    for (int j = 0; j < HIDDEN / 32; ++j) {
        const int c = lane + 32 * j;
        atomicAdd(&ag[c], alpha * hs[c]);
    }
}

// ---------------------------------------------------------------------------
static inline int cdiv(int a, int b) { return (a + b - 1) / b; }

extern "C" void kernel_launch(void* const* d_in, const int* in_sizes, int n_in,
                              void* d_out, int out_size, void* d_ws, size_t ws_size,
                              hipStream_t stream)
{
    const float* nf    = (const float*)d_in[0];
    const int*   ei    = (const int*)  d_in[1];   // [2, E] (int32 under JAX default x64=off)
    const float* W0    = (const float*)d_in[2];   // [256, 128]
    const float* b0    = (const float*)d_in[3];   // [256]
    const float* Wlin  = (const float*)d_in[4];   // [3, 256, 512]
    const float* Wattn = (const float*)d_in[5];   // [3, 512]

    const int nNodes = in_sizes[0] / NF_IN;       // 10000
    const int E      = in_sizes[1] / 2;           // 320000
    const int* src = ei;
    const int* dst = ei + E;

    // ---- workspace partition (256B aligned) ----
    char* ws = (char*)d_ws;
    size_t off = 0;
    auto alloc = [&](size_t bytes) {
        void* p = ws + off;
        off += (bytes + 255) & ~(size_t)255;
        return p;
    };
    float*    hA     = (float*)   alloc((size_t)nNodes * HIDDEN * 4);
    float*    hB     = (float*)   alloc((size_t)nNodes * HIDDEN * 4);
    float*    agg    = (float*)   alloc((size_t)nNodes * HIDDEN * 4);
    float*    asrc   = (float*)   alloc((size_t)nNodes * 4);
    float*    adst   = (float*)   alloc((size_t)nNodes * 4);
    float*    denom  = (float*)   alloc((size_t)nNodes * 4);
    unsigned* segmax = (unsigned*)alloc((size_t)nNodes * 4);
    float*    ef     = (float*)   alloc((size_t)E * 4);
    float*    efexp  = (float*)   alloc((size_t)E * 4);
    (void)ws_size; (void)n_in; (void)out_size;

    const dim3 blk(256);
    const int wavesPerBlk = 256 / 32;

    // ---- h = nf @ W0^T + b0 ----
    {
        const int nTiles = (nNodes / 16) * (HIDDEN / 64);
        gemm_atb_wmma_kernel<<<cdiv(nTiles, wavesPerBlk), blk, 0, stream>>>(
            nf, nullptr, NF_IN, NF_IN, NF_IN, W0, b0, hA, nNodes, HIDDEN, 0);
    }

    float* hcur = hA;
    for (int layer = 0; layer < DEPTH; ++layer) {
        const float* Wl = Wlin  + (size_t)layer * HIDDEN * (2 * HIDDEN);
        const float* Wa = Wattn + (size_t)layer * (2 * HIDDEN);
        float* hnext = (layer == DEPTH - 1) ? (float*)d_out
                                            : ((hcur == hA) ? hB : hA);

        // reset segment state + accumulator
        fill_u32_kernel<<<cdiv(nNodes, 256), blk, 0, stream>>>(segmax, ENC_NEGINF, nNodes);
        fill_f32_kernel<<<cdiv(nNodes, 256), blk, 0, stream>>>(denom, 0.0f, nNodes);
        fill_f32_kernel<<<cdiv(nNodes * HIDDEN, 256), blk, 0, stream>>>(agg, 0.0f, nNodes * HIDDEN);

        // per-node attention projections
        att_node_kernel<<<cdiv(nNodes, wavesPerBlk), blk, 0, stream>>>(
            hcur, Wa, asrc, adst, nNodes);

        // segment softmax over edges grouped by dst
        edge_logit_max_kernel<<<cdiv(E, 256), blk, 0, stream>>>(
            src, dst, asrc, adst, ef, segmax, E);
        edge_exp_sum_kernel<<<cdiv(E, 256), blk, 0, stream>>>(
            dst, ef, segmax, efexp, denom, E);

        // weighted aggregation (one wave per edge)
        edge_aggregate_kernel<<<cdiv(E, wavesPerBlk), blk, 0, stream>>>(
            src, dst, hcur, efexp, denom, agg, E);

        // h' = relu([h | agg] @ Wl^T)
        const int nTiles = (nNodes / 16) * (HIDDEN / 64);
        gemm_atb_wmma_kernel<<<cdiv(nTiles, wavesPerBlk), blk, 0, stream>>>(
            hcur, agg, HIDDEN, 2 * HIDDEN, HIDDEN, Wl, nullptr, hnext,
            nNodes, HIDDEN, 1);

        hcur = hnext;
    }
}